// GraphGenerator_55542517072529
// MI455X (gfx1250) — compile-verified
//
#include <hip/hip_runtime.h>
#include <math.h>

typedef float v2f __attribute__((ext_vector_type(2)));
typedef float v8f __attribute__((ext_vector_type(8)));

#define HIDC 128
#define KSEL 512
#define NB 2
#define CAP 4096
#define NEG_INF (-__builtin_huge_valf())

__device__ __forceinline__ float gelu_exact(float x) {
    return 0.5f * x * (1.0f + erff(x * 0.70710678118654752440f));
}

// ---------------------------------------------------------------------------
// Kernel 1: reliability head over the full map.
// rel[b,p] = rw2 . gelu(rw1 @ X[:,p] + rb1) + rb2, with border -> -inf.
// One block = 128 pixels, 8 waves; each wave: 16 pixels x 128 outputs via
// 8 accumulator tiles of v_wmma_f32_16x16x4_f32 (K-loop of 32 steps).
// Operands fetched as lane-addressed float2 LDS loads (broadcast across the
// two lane halves, zero cndmask glue between WMMAs).
// ---------------------------------------------------------------------------
__global__ __launch_bounds__(256, 1)
void k_rel_head(const float* __restrict__ fm, const float* __restrict__ rw1,
                const float* __restrict__ rb1, const float* __restrict__ rw2,
                const float* __restrict__ rb2, float* __restrict__ rel,
                int H, int W, int border)
{
    __shared__ __align__(16) float Wlds[128 * 132];   // [o][c], padded row
    __shared__ __align__(16) float Xlds[128 * 132];   // [pixel][c], padded row
    __shared__ float rb1s[128];
    __shared__ float rw2s[128];

    const int HW = H * W;
    const int b = blockIdx.y;
    const int pbase = blockIdx.x * 128;
    const int t = threadIdx.x;

    for (int li = t; li < 128 * 128; li += 256)
        Wlds[(li >> 7) * 132 + (li & 127)] = rw1[li];
    if (t < 128) { rb1s[t] = rb1[t]; rw2s[t] = rw2[t]; }
    {
        const float* src = fm + (size_t)b * HIDC * HW + pbase;
        int p = t & 127;
        for (int c = (t >> 7); c < 128; c += 2)
            Xlds[p * 132 + c] = src[(size_t)c * HW + p];
    }
    __syncthreads();

    const int lane = t & 31;
    const int wv = t >> 5;
    const int lp = lane & 15;
    const int hi = lane >> 4;            // 0: K/K+1 rows, 1: K+2/K+3 rows
    const int pcol = wv * 16 + lp;       // pixel column within tile

    v8f acc[8];
    #pragma unroll
    for (int ob = 0; ob < 8; ++ob)
        #pragma unroll
        for (int e = 0; e < 8; ++e) acc[ob][e] = 0.0f;

    const float* xrow = &Xlds[pcol * 132 + hi * 2];
    const float* wrow = &Wlds[lp * 132 + hi * 2];

    for (int k = 0; k < 128; k += 4) {
        const v2f B = *(const v2f*)(xrow + k);
        #pragma unroll
        for (int ob = 0; ob < 8; ++ob) {
            const v2f A = *(const v2f*)(wrow + ob * (16 * 132) + k);
            acc[ob] = __builtin_amdgcn_wmma_f32_16x16x4_f32(
                false, A, false, B, (short)0, acc[ob], false, false);
        }
    }

    // bias + exact GELU + rw2 dot (second conv to 1 channel)
    float partial = 0.0f;
    #pragma unroll
    for (int ob = 0; ob < 8; ++ob)
        #pragma unroll
        for (int v = 0; v < 8; ++v) {
            int o = ob * 16 + v + hi * 8;
            float g = gelu_exact(acc[ob][v] + rb1s[o]);
            partial += rw2s[o] * g;
        }
    partial += __shfl_xor(partial, 16, 32);
    if (hi == 0) {
        int pix = pbase + pcol;
        int x = pix % W, y = pix / W;
        float r = partial + rb2[0];
        bool inb = (x >= border) && (x < W - border) &&
                   (y >= border) && (y < H - border);
        rel[(size_t)b * HW + pix] = inb ? r : NEG_INF;
    }
}

// ---------------------------------------------------------------------------
// Separable max-pool (padding behaves like -inf since window is clamped).
// ---------------------------------------------------------------------------
__global__ void k_maxpool_h(const float* __restrict__ in, float* __restrict__ out,
                            int H, int W, int r) {
    int i = blockIdx.x * blockDim.x + threadIdx.x;
    if (i >= NB * H * W) return;
    int x = i % W;
    int rowbase = i - x;
    int x0 = max(0, x - r), x1 = min(W - 1, x + r);
    float m = NEG_INF;
    for (int xx = x0; xx <= x1; ++xx) m = fmaxf(m, in[rowbase + xx]);
    out[i] = m;
}

__global__ void k_maxpool_v(const float* __restrict__ in, float* __restrict__ out,
                            int H, int W, int r) {
    int i = blockIdx.x * blockDim.x + threadIdx.x;
    int hw = H * W;
    if (i >= NB * hw) return;
    int p = i % hw;
    int x = p % W, y = p / W;
    int base = i - p;
    int y0 = max(0, y - r), y1 = min(H - 1, y + r);
    float m = NEG_INF;
    for (int yy = y0; yy <= y1; ++yy) m = fmaxf(m, in[base + yy * W + x]);
    out[i] = m;
}

__global__ void k_eqmask(const float* __restrict__ w, const float* __restrict__ mp,
                         float* __restrict__ mask, int n) {
    int i = blockIdx.x * blockDim.x + threadIdx.x;
    if (i < n) mask[i] = (w[i] == mp[i]) ? 1.0f : 0.0f;
}

__global__ void k_suppress(const float* __restrict__ maskpool, const float* __restrict__ w,
                           float* __restrict__ supp, float* __restrict__ suppw, int n) {
    int i = blockIdx.x * blockDim.x + threadIdx.x;
    if (i >= n) return;
    bool s = maskpool[i] > 0.0f;
    supp[i] = s ? 1.0f : 0.0f;
    suppw[i] = s ? NEG_INF : w[i];
}

__global__ void k_update_mask(const float* __restrict__ suppw, const float* __restrict__ mpsw,
                              const float* __restrict__ supp, float* __restrict__ mask, int n) {
    int i = blockIdx.x * blockDim.x + threadIdx.x;
    if (i >= n) return;
    bool newmax = (suppw[i] == mpsw[i]);
    bool s = supp[i] > 0.0f;
    mask[i] = ((mask[i] > 0.0f) || (newmax && !s)) ? 1.0f : 0.0f;
}

__global__ void k_final_nms(float* __restrict__ w, const float* __restrict__ mask, int n) {
    int i = blockIdx.x * blockDim.x + threadIdx.x;
    if (i < n) w[i] = (mask[i] > 0.0f) ? w[i] : NEG_INF;
}

// ---------------------------------------------------------------------------
// Top-K: compact survivors as 64-bit keys, bitonic sort per (b,scale).
// key = sortable(value) << 32 | (0xFFFFFFFF - idx)  -> desc value, asc idx.
// ---------------------------------------------------------------------------
__global__ void k_zero_cnt(int* cnt) { if (threadIdx.x < NB) cnt[threadIdx.x] = 0; }

__global__ void k_compact(const float* __restrict__ rel, unsigned long long* __restrict__ keys,
                          int* __restrict__ cnt, int HW) {
    int i = blockIdx.x * blockDim.x + threadIdx.x;
    if (i >= NB * HW) return;
    int b = i / HW;
    int pix = i - b * HW;
    float v = rel[i];
    if (v != NEG_INF) {
        int pos = atomicAdd(&cnt[b], 1);
        if (pos < CAP) {
            unsigned u = __float_as_uint(v);
            unsigned s = (u & 0x80000000u) ? ~u : (u | 0x80000000u);
            keys[(size_t)b * CAP + pos] =
                ((unsigned long long)s << 32) | (unsigned long long)(0xFFFFFFFFu - (unsigned)pix);
        }
    }
}

__global__ __launch_bounds__(1024, 1)
void k_topk(const unsigned long long* __restrict__ keys, const int* __restrict__ cnt,
            float* __restrict__ out, int* __restrict__ kpidx, int H, int W, int scale) {
    __shared__ unsigned long long lds[CAP];
    const int b = blockIdx.x;
    const int t = threadIdx.x;
    const int n = min(cnt[b], CAP);

    for (int i = t; i < CAP; i += 1024) {
        if (i < n) lds[i] = keys[(size_t)b * CAP + i];
        else {
            unsigned s = ~__float_as_uint(NEG_INF);       // sortable(-inf)
            unsigned idx = (unsigned)(i - n);             // jax pads with lowest indices
            lds[i] = ((unsigned long long)s << 32) | (unsigned long long)(0xFFFFFFFFu - idx);
        }
    }
    __syncthreads();

    for (int size = 2; size <= CAP; size <<= 1) {
        for (int stride = size >> 1; stride > 0; stride >>= 1) {
            for (int i = t; i < CAP; i += 1024) {
                int j = i ^ stride;
                if (j > i) {
                    unsigned long long a = lds[i], c = lds[j];
                    bool desc = (i & size) == 0;
                    if (desc ? (a < c) : (a > c)) { lds[i] = c; lds[j] = a; }
                }
            }
            __syncthreads();
        }
    }

    if (t < KSEL) {
        unsigned long long kk = lds[t];
        unsigned s = (unsigned)(kk >> 32);
        unsigned u = (s & 0x80000000u) ? (s & 0x7FFFFFFFu) : ~s;
        float val = __uint_as_float(u);
        int idx = (int)(0xFFFFFFFFu - (unsigned)(kk & 0xFFFFFFFFull));
        int x = idx % W, y = idx / W;
        float nx = (float)x / (float)(W - 1) * 2.0f - 1.0f;
        float ny = (float)y / (float)(H - 1) * 2.0f - 1.0f;
        int row = scale * KSEL + t;
        out[((size_t)b * 2048 + row) * 2 + 0] = nx;
        out[((size_t)b * 2048 + row) * 2 + 1] = ny;
        out[8192 + (size_t)b * 2048 + row] = val;
        kpidx[b * KSEL + t] = idx;
    }
}

// ---------------------------------------------------------------------------
// Feature head at the 4*K gathered corner pixels (2048 columns per batch).
// GEMM1 (global-gathered B) -> GELU -> LDS transpose -> GEMM2, fp32 WMMA.
// ---------------------------------------------------------------------------
__global__ __launch_bounds__(256, 1)
void k_feat(const float* __restrict__ fm, const float* __restrict__ fw1,
            const float* __restrict__ fb1, const float* __restrict__ fw2,
            const float* __restrict__ fb2, const int* __restrict__ kpidx,
            float* __restrict__ cfeat, int H, int W)
{
    __shared__ __align__(16) float Wlds[128 * 132];
    __shared__ __align__(16) float Hlds[128 * 132];   // hidden [col][o]
    __shared__ int   pixc[128];
    __shared__ float fb1s[128];
    __shared__ float fb2s[128];

    const int b = blockIdx.y, ct = blockIdx.x, t = threadIdx.x;
    const int HW = H * W;

    if (t < 128) {
        int col = ct * 128 + t;
        int kp = col >> 2, corner = col & 3;
        int idx = kpidx[b * KSEL + kp];
        int xq = idx % W, yq = idx / W;
        float nx = (float)xq / (float)(W - 1) * 2.0f - 1.0f;
        float ny = (float)yq / (float)(H - 1) * 2.0f - 1.0f;
        float ix = ((nx + 1.0f) * (float)W - 1.0f) * 0.5f;
        float iy = ((ny + 1.0f) * (float)H - 1.0f) * 0.5f;
        float x0 = floorf(ix), y0 = floorf(iy);
        float cx = (corner & 1) ? (x0 + 1.0f) : x0;
        float cy = (corner & 2) ? (y0 + 1.0f) : y0;
        int xi = (int)fminf(fmaxf(cx, 0.0f), (float)(W - 1));
        int yi = (int)fminf(fmaxf(cy, 0.0f), (float)(H - 1));
        pixc[t] = yi * W + xi;
        fb1s[t] = fb1[t]; fb2s[t] = fb2[t];
    }
    for (int li = t; li < 128 * 128; li += 256)
        Wlds[(li >> 7) * 132 + (li & 127)] = fw1[li];
    __syncthreads();

    const int lane = t & 31, wv = t >> 5, lp = lane & 15, hi = lane >> 4;
    const int pcol = wv * 16 + lp;
    const float* fmb = fm + (size_t)b * HIDC * HW;
    const int pix = pixc[pcol];
    const float* wrow = &Wlds[lp * 132 + hi * 2];

    v8f acc[8];
    #pragma unroll
    for (int ob = 0; ob < 8; ++ob)
        #pragma unroll
        for (int e = 0; e < 8; ++e) acc[ob][e] = 0.0f;

    for (int k = 0; k < 128; k += 4) {
        int c0 = k + hi * 2;
        v2f B;
        B.x = fmb[(size_t)c0 * HW + pix];
        B.y = fmb[(size_t)(c0 + 1) * HW + pix];
        #pragma unroll
        for (int ob = 0; ob < 8; ++ob) {
            const v2f A = *(const v2f*)(wrow + ob * (16 * 132) + k);
            acc[ob] = __builtin_amdgcn_wmma_f32_16x16x4_f32(
                false, A, false, B, (short)0, acc[ob], false, false);
        }
    }
    __syncthreads();   // done reading fw1 from Wlds

    // hidden = gelu(acc + fb1) -> Hlds[col][o], written as float4 chunks
    #pragma unroll
    for (int ob = 0; ob < 8; ++ob) {
        float4 g0, g1;
        g0.x = gelu_exact(acc[ob][0] + fb1s[ob * 16 + 0 + hi * 8]);
        g0.y = gelu_exact(acc[ob][1] + fb1s[ob * 16 + 1 + hi * 8]);
        g0.z = gelu_exact(acc[ob][2] + fb1s[ob * 16 + 2 + hi * 8]);
        g0.w = gelu_exact(acc[ob][3] + fb1s[ob * 16 + 3 + hi * 8]);
        g1.x = gelu_exact(acc[ob][4] + fb1s[ob * 16 + 4 + hi * 8]);
        g1.y = gelu_exact(acc[ob][5] + fb1s[ob * 16 + 5 + hi * 8]);
        g1.z = gelu_exact(acc[ob][6] + fb1s[ob * 16 + 6 + hi * 8]);
        g1.w = gelu_exact(acc[ob][7] + fb1s[ob * 16 + 7 + hi * 8]);
        float* hdst = &Hlds[pcol * 132 + ob * 16 + hi * 8];
        *(float4*)(hdst + 0) = g0;
        *(float4*)(hdst + 4) = g1;
    }
    for (int li = t; li < 128 * 128; li += 256)
        Wlds[(li >> 7) * 132 + (li & 127)] = fw2[li];
    __syncthreads();

    v8f acc2[8];
    #pragma unroll
    for (int ob = 0; ob < 8; ++ob)
        #pragma unroll
        for (int e = 0; e < 8; ++e) acc2[ob][e] = 0.0f;

    const float* hrow = &Hlds[pcol * 132 + hi * 2];
    for (int k = 0; k < 128; k += 4) {
        const v2f B = *(const v2f*)(hrow + k);
        #pragma unroll
        for (int ob = 0; ob < 8; ++ob) {
            const v2f A = *(const v2f*)(wrow + ob * (16 * 132) + k);
            acc2[ob] = __builtin_amdgcn_wmma_f32_16x16x4_f32(
                false, A, false, B, (short)0, acc2[ob], false, false);
        }
    }

    float* dst = cfeat + ((size_t)b * 2048 + (size_t)(ct * 128 + pcol)) * 128;
    #pragma unroll
    for (int ob = 0; ob < 8; ++ob) {
        float4 f0, f1;
        f0.x = acc2[ob][0] + fb2s[ob * 16 + 0 + hi * 8];
        f0.y = acc2[ob][1] + fb2s[ob * 16 + 1 + hi * 8];
        f0.z = acc2[ob][2] + fb2s[ob * 16 + 2 + hi * 8];
        f0.w = acc2[ob][3] + fb2s[ob * 16 + 3 + hi * 8];
        f1.x = acc2[ob][4] + fb2s[ob * 16 + 4 + hi * 8];
        f1.y = acc2[ob][5] + fb2s[ob * 16 + 5 + hi * 8];
        f1.z = acc2[ob][6] + fb2s[ob * 16 + 6 + hi * 8];
        f1.w = acc2[ob][7] + fb2s[ob * 16 + 7 + hi * 8];
        *(float4*)(dst + ob * 16 + hi * 8 + 0) = f0;
        *(float4*)(dst + ob * 16 + hi * 8 + 4) = f1;
    }
}

// ---------------------------------------------------------------------------
// Bilinear combine of the 4 corner features (zero invalid corners).
// ---------------------------------------------------------------------------
__global__ void k_combine(const float* __restrict__ cfeat, const int* __restrict__ kpidx,
                          float* __restrict__ out, int H, int W, int scale) {
    int i = blockIdx.x * blockDim.x + threadIdx.x;
    if (i >= NB * KSEL * 128) return;
    int o = i & 127;
    int kp = (i >> 7) % KSEL;
    int b = (i >> 7) / KSEL;
    int idx = kpidx[b * KSEL + kp];
    int xq = idx % W, yq = idx / W;
    float nx = (float)xq / (float)(W - 1) * 2.0f - 1.0f;
    float ny = (float)yq / (float)(H - 1) * 2.0f - 1.0f;
    float ix = ((nx + 1.0f) * (float)W - 1.0f) * 0.5f;
    float iy = ((ny + 1.0f) * (float)H - 1.0f) * 0.5f;
    float x0 = floorf(ix), y0 = floorf(iy);
    float wx1 = ix - x0, wy1 = iy - y0;
    float wgt[4] = { (1.0f - wx1) * (1.0f - wy1), wx1 * (1.0f - wy1),
                     (1.0f - wx1) * wy1,          wx1 * wy1 };
    float accv = 0.0f;
    #pragma unroll
    for (int c = 0; c < 4; ++c) {
        float cx = (c & 1) ? (x0 + 1.0f) : x0;
        float cy = (c & 2) ? (y0 + 1.0f) : y0;
        bool valid = (cx >= 0.0f) && (cx <= (float)(W - 1)) &&
                     (cy >= 0.0f) && (cy <= (float)(H - 1));
        if (valid)
            accv += wgt[c] * cfeat[((size_t)b * 2048 + (size_t)(kp * 4 + c)) * 128 + o];
    }
    out[12288 + ((size_t)b * 2048 + (size_t)(scale * KSEL + kp)) * 128 + o] = accv;
}

// ---------------------------------------------------------------------------
extern "C" void kernel_launch(void* const* d_in, const int* in_sizes, int n_in,
                              void* d_out, int out_size, void* d_ws, size_t ws_size,
                              hipStream_t stream) {
    const float* fms[4] = { (const float*)d_in[0], (const float*)d_in[1],
                            (const float*)d_in[2], (const float*)d_in[3] };
    const float* rw1 = (const float*)d_in[4];
    const float* rb1 = (const float*)d_in[5];
    const float* rw2 = (const float*)d_in[6];
    const float* rb2 = (const float*)d_in[7];
    const float* fw1 = (const float*)d_in[8];
    const float* fb1 = (const float*)d_in[9];
    const float* fw2 = (const float*)d_in[10];
    const float* fb2 = (const float*)d_in[11];
    float* out = (float*)d_out;

    const size_t MAPF = (size_t)NB * 512 * 512;   // floats per map buffer
    char* ws = (char*)d_ws;
    float* REL   = (float*)ws;                    ws += MAPF * 4;
    float* T0    = (float*)ws;                    ws += MAPF * 4;
    float* T1    = (float*)ws;                    ws += MAPF * 4;
    float* MASK  = (float*)ws;                    ws += MAPF * 4;
    float* SUPP  = (float*)ws;                    ws += MAPF * 4;
    float* SUPPW = (float*)ws;                    ws += MAPF * 4;
    float* CFEAT = (float*)ws;                    ws += (size_t)NB * 2048 * 128 * 4;
    unsigned long long* KEYS = (unsigned long long*)ws; ws += (size_t)NB * CAP * 8;
    int* CNT   = (int*)ws;                        ws += 64;
    int* KPIDX = (int*)ws;                        ws += (size_t)NB * KSEL * 4;

    for (int s = 0; s < 4; ++s) {
        const int H = 512 >> s, W = 512 >> s, HW = H * W;
        const int r = max(1, H / 64);             // radius == border
        const int n = NB * HW;
        const int eb = (n + 255) / 256;

        k_rel_head<<<dim3(HW / 128, NB), 256, 0, stream>>>(
            fms[s], rw1, rb1, rw2, rb2, REL, H, W, r);

        // simple_nms: mp(rel), then 2 suppress iterations
        k_maxpool_h<<<eb, 256, 0, stream>>>(REL, T0, H, W, r);
        k_maxpool_v<<<eb, 256, 0, stream>>>(T0, T1, H, W, r);
        k_eqmask<<<eb, 256, 0, stream>>>(REL, T1, MASK, n);
        for (int it = 0; it < 2; ++it) {
            k_maxpool_h<<<eb, 256, 0, stream>>>(MASK, T0, H, W, r);
            k_maxpool_v<<<eb, 256, 0, stream>>>(T0, T1, H, W, r);
            k_suppress<<<eb, 256, 0, stream>>>(T1, REL, SUPP, SUPPW, n);
            k_maxpool_h<<<eb, 256, 0, stream>>>(SUPPW, T0, H, W, r);
            k_maxpool_v<<<eb, 256, 0, stream>>>(T0, T1, H, W, r);
            k_update_mask<<<eb, 256, 0, stream>>>(SUPPW, T1, SUPP, MASK, n);
        }
        k_final_nms<<<eb, 256, 0, stream>>>(REL, MASK, n);

        k_zero_cnt<<<1, 32, 0, stream>>>(CNT);
        k_compact<<<eb, 256, 0, stream>>>(REL, KEYS, CNT, HW);
        k_topk<<<NB, 1024, 0, stream>>>(KEYS, CNT, out, KPIDX, H, W, s);

        k_feat<<<dim3(16, NB), 256, 0, stream>>>(
            fms[s], fw1, fb1, fw2, fb2, KPIDX, CFEAT, H, W);
        k_combine<<<(NB * KSEL * 128 + 255) / 256, 256, 0, stream>>>(
            CFEAT, KPIDX, out, H, W, s);
    }
}